// LightGCNConv_18605798326906
// MI455X (gfx1250) — compile-verified
//
#include <hip/hip_runtime.h>
#include <hip/hip_bf16.h>

#define AS1 __attribute__((address_space(1)))
#define AS3 __attribute__((address_space(3)))

#define TILE 256            // edges per tile
#define DIM  64             // embedding dim
#define LANES_PER_EDGE 16   // 16 lanes x float4 = 64 floats = one E row

typedef int v4i __attribute__((ext_vector_type(4)));

// generic->LDS addrspacecast on AMDGPU is a trunc to the low 32 bits, so the
// uint32_t round-trip is value-correct for __shared__ data; global pointers
// keep their flat value in AS(1).
__device__ __forceinline__ void async_copy_b128(const void* g, void* l) {
    __builtin_amdgcn_global_load_async_to_lds_b128(
        (AS1 v4i*)(uintptr_t)g,
        (AS3 v4i*)(uint32_t)(uintptr_t)l,
        /*offset=*/0, /*cpol=*/0);
}

__global__ __launch_bounds__(256) void lightgcn_spmm_kernel(
    const int*   __restrict__ edge_src,
    const int*   __restrict__ edge_dst,
    const float* __restrict__ edge_val,
    const float* __restrict__ E,
    float*       __restrict__ out,
    int n_edges)
{
    // Double-buffered edge-tile staging (2 x 3 KB).
    __shared__ __align__(16) int   sh_src[2][TILE];
    __shared__ __align__(16) int   sh_dst[2][TILE];
    __shared__ __align__(16) float sh_val[2][TILE];

    const int tid     = threadIdx.x;
    const int n_tiles = (n_edges + TILE - 1) / TILE;
    const int stride  = (int)gridDim.x;

    // --- tile fillers ------------------------------------------------------
    // Full tile: async DMA, 64 threads x b128 per stream (ASYNCcnt tracked).
    auto issue_async = [&](int base, int buf) {
        if (tid < 192) {
            const int which = tid >> 6;        // 0=src 1=dst 2=val
            const int slot  = (tid & 63) * 4;  // dword index within tile
            const void* gp;
            void*       lp;
            if (which == 0)      { gp = edge_src + base + slot; lp = &sh_src[buf][slot]; }
            else if (which == 1) { gp = edge_dst + base + slot; lp = &sh_dst[buf][slot]; }
            else                 { gp = edge_val + base + slot; lp = &sh_val[buf][slot]; }
            async_copy_b128(gp, lp);
        }
    };
    // Partial tail tile: plain loads + LDS stores (DScnt tracked; the
    // compiler inserts the dscnt wait before the barrier).
    auto fill_sync = [&](int base, int n, int buf) {
        if (tid < n) {
            sh_src[buf][tid] = edge_src[base + tid];
            sh_dst[buf][tid] = edge_dst[base + tid];
            sh_val[buf][tid] = edge_val[base + tid];
        }
    };

    int t = (int)blockIdx.x;
    if (t >= n_tiles) return;

    // Prologue: stage the first tile into buffer 0.
    if (n_edges - t * TILE >= TILE) issue_async(t * TILE, 0);
    else                            fill_sync(t * TILE, n_edges - t * TILE, 0);

    const int group = tid >> 4;   // 0..15: which edge of the pass
    const int sub   = tid & 15;   // float4 slice of the 64-float row

    int parity = 0;
    for (; t < n_tiles; t += stride, parity ^= 1) {
        // Software pipeline: issue tile t+stride's DMA into the other buffer,
        // then wait only for the OLDEST outstanding async op (tile t) —
        // ASYNCcnt completes in order, so <=1 means "current tile landed,
        // next tile still in flight".
        const int tnext = t + stride;
        if (tnext < n_tiles && n_edges - tnext * TILE >= TILE) {
            issue_async(tnext * TILE, parity ^ 1);
            __builtin_amdgcn_s_wait_asynccnt(1);
        } else {
            if (tnext < n_tiles)
                fill_sync(tnext * TILE, n_edges - tnext * TILE, parity ^ 1);
            __builtin_amdgcn_s_wait_asynccnt(0);
        }
        __syncthreads();   // all waves' copies of tile t visible

        // 16 groups of 16 lanes; each group handles one edge per pass.
        // A wave32 streams two E-rows (2 x 256B) per pass: fully coalesced
        // b128 gathers out of L2 (E = 25.6 MB << 192 MB L2).
        const int tile_n = min(TILE, n_edges - t * TILE);
        #pragma unroll 4
        for (int p = 0; p < TILE / LANES_PER_EDGE; ++p) {
            const int idx = p * 16 + group;
            if (idx < tile_n) {
                const int   s = sh_src[parity][idx];   // LDS broadcast
                const int   d = sh_dst[parity][idx];
                const float w = sh_val[parity][idx];

                const float4 v = *reinterpret_cast<const float4*>(
                    E + (size_t)s * DIM + (size_t)sub * 4);

                float* o = out + (size_t)d * DIM + (size_t)sub * 4;
                // Non-returning f32 adds -> L2 atomic units (out L2-resident).
                atomicAdd(o + 0, v.x * w);
                atomicAdd(o + 1, v.y * w);
                atomicAdd(o + 2, v.z * w);
                atomicAdd(o + 3, v.w * w);
            }
        }
        __syncthreads();   // done reading buf[parity] before it is re-DMAed
    }
}

extern "C" void kernel_launch(void* const* d_in, const int* in_sizes, int n_in,
                              void* d_out, int out_size, void* d_ws, size_t ws_size,
                              hipStream_t stream) {
    const int*   edge_src = (const int*)  d_in[0];
    const int*   edge_dst = (const int*)  d_in[1];
    const float* edge_val = (const float*)d_in[2];
    const float* E        = (const float*)d_in[3];
    float*       out      = (float*)d_out;
    const int n_edges = in_sizes[0];

    // Atomic accumulation requires a zeroed output every call (graph-safe).
    (void)hipMemsetAsync(d_out, 0, (size_t)out_size * sizeof(float), stream);

    const int n_tiles = (n_edges + TILE - 1) / TILE;
    const int blocks  = n_tiles < 4096 ? n_tiles : 4096;
    lightgcn_spmm_kernel<<<blocks, 256, 0, stream>>>(
        edge_src, edge_dst, edge_val, E, out, n_edges);
}